// SimpleMLP_55628416417846
// MI455X (gfx1250) — compile-verified
//
#include <hip/hip_runtime.h>
#include <stdint.h>

// ---------------------------------------------------------------------------
// SimpleMLP (k-winners + sparse-masked linear) for MI455X (gfx1250, wave32).
// Memory-bound problem (~0.9 GB min traffic @ 23.3 TB/s ~= 39 us); f32 WMMA
// (V_WMMA_F32_16X16X4_F32) keeps reference precision while staying far below
// the HBM roof. Activations (154 MB) are kept L2-resident; weight+mask stream
// with non-temporal loads.
// ---------------------------------------------------------------------------

typedef float v2f __attribute__((ext_vector_type(2)));
typedef float v4f __attribute__((ext_vector_type(4)));
typedef float v8f __attribute__((ext_vector_type(8)));

#define B_DIM   256
#define IN_F    150528          // 3*224*224
#define N_CLS   1000
#define TOPK    (IN_F / 10)     // int(0.1 * 150528) = 15052
#define SPLITK  8
#define KC      (IN_F / SPLITK) // 18816, divisible by KT
#define KT      32              // K tile staged in LDS per iteration
#define MT      128             // workgroup M tile
#define NTILE   128             // workgroup N tile
#define NT_PAD  1024            // padded N stride for split-K partials
#define LDSK    (KT + 4)        // 36 floats -> 144B row stride (16B aligned)

// order-preserving float <-> uint mapping (descending radix select)
__device__ __forceinline__ uint32_t f2ord(float f) {
    uint32_t u = __float_as_uint(f);
    return (u & 0x80000000u) ? ~u : (u | 0x80000000u);
}
__device__ __forceinline__ float ord2f(uint32_t u) {
    uint32_t b = (u & 0x80000000u) ? (u & 0x7FFFFFFFu) : ~u;
    return __uint_as_float(b);
}

// -------------------------------------------------------------------------
// Kernel 1: exact per-row k-th largest via 4-round byte radix select.
// One block per batch row; LDS histogram with ds_add_u32 atomics.
// -------------------------------------------------------------------------
__global__ __launch_bounds__(256) void topk_thresh_kernel(
    const float* __restrict__ x, float* __restrict__ thresh)
{
    __shared__ uint32_t hist[256];
    __shared__ uint32_t sPrefix, sWant;
    const int row = blockIdx.x;
    const float* xr = x + (size_t)row * IN_F;

    if (threadIdx.x == 0) { sPrefix = 0u; sWant = (uint32_t)TOPK; }

    for (int pass = 3; pass >= 0; --pass) {
        hist[threadIdx.x] = 0u;
        __syncthreads();
        const uint32_t prefix = sPrefix;
        const int shift = pass * 8;
        for (int i = threadIdx.x; i < IN_F; i += 256) {
            uint32_t u = f2ord(xr[i]);
            bool match = (pass == 3) ||
                         ((u >> (shift + 8)) == (prefix >> (shift + 8)));
            if (match) atomicAdd(&hist[(u >> shift) & 0xFFu], 1u);
        }
        __syncthreads();
        if (threadIdx.x == 0) {
            uint32_t want = sWant, cum = 0u;
            for (int bin = 255; bin >= 0; --bin) {
                cum += hist[bin];
                if (cum >= want) {
                    sWant   = want - (cum - hist[bin]);   // rank inside bin
                    sPrefix = prefix | ((uint32_t)bin << shift);
                    break;
                }
            }
        }
        __syncthreads();
    }
    if (threadIdx.x == 0) thresh[row] = ord2f(sPrefix);
}

// -------------------------------------------------------------------------
// Kernel 2: split-K WMMA GEMM.  out_partial[ks] = act_tile @ (w*mask)^T.
// Workgroup: 256 threads = 8 waves; tile 128(M) x 128(N); wave tile 32x64
// => 2x4 accumulators of v_wmma_f32_16x16x4_f32.
// A-tile thresholding fused into the global->LDS stage.
// -------------------------------------------------------------------------
__global__ __launch_bounds__(256) void wmma_gemm_kernel(
    const float* __restrict__ x, const float* __restrict__ w,
    const unsigned char* __restrict__ mask, const float* __restrict__ thresh,
    float* __restrict__ partial)
{
    __shared__ float Alds[MT][LDSK];
    __shared__ float Blds[NTILE][LDSK];

    const int tid  = threadIdx.x;
    const int lane = tid & 31;
    const int wave = tid >> 5;      // 0..7
    const int wm   = wave >> 1;     // 0..3 : M sub-block of 32 rows
    const int wn   = wave & 1;      // 0..1 : N sub-block of 64 cols
    const int hi   = lane >> 4;     // lane half (K split of fragments)
    const int l15  = lane & 15;

    const int bn = blockIdx.x;      // 0..7  N tile (covers 1024 >= 1000)
    const int bm = blockIdx.y;      // 0..1  M tile
    const int ks = blockIdx.z;      // 0..SPLITK-1
    const size_t kbase = (size_t)ks * KC;

    // Per-thread fixed load slots: slot s in [0,1024): row = s>>3, col4 = (s&7)*4
    int   arow[4];
    float ath[4];
    const float* gA[4];
    const float* gW[4];
    const uint32_t* gM[4];
    bool nvalid[4];
#pragma unroll
    for (int i = 0; i < 4; ++i) {
        const int s = tid + i * 256;
        const int r = s >> 3;
        const int c = (s & 7) << 2;
        arow[i] = r;
        ath[i]  = thresh[bm * MT + r];
        gA[i]   = x + (size_t)(bm * MT + r) * IN_F + kbase + c;
        const int n = bn * NTILE + r;
        nvalid[i] = (n < N_CLS);
        const size_t woff = (size_t)(nvalid[i] ? n : 0) * IN_F + kbase + c;
        gW[i] = w + woff;
        gM[i] = (const uint32_t*)(mask + woff);
    }

    v8f acc[2][4];
#pragma unroll
    for (int mi = 0; mi < 2; ++mi)
#pragma unroll
        for (int ni = 0; ni < 4; ++ni) acc[mi][ni] = {};

    for (int kt = 0; kt < KC; kt += KT) {
        // ---- stage A (thresholded activations) and B (masked weights) ----
#pragma unroll
        for (int i = 0; i < 4; ++i) {
            const int s = tid + i * 256;
            const int r = s >> 3;
            const int c = (s & 7) << 2;

            v4f av = *(const v4f*)(gA[i] + kt);
            const float th = ath[i];
            av.x = (av.x >= th) ? av.x : 0.0f;
            av.y = (av.y >= th) ? av.y : 0.0f;
            av.z = (av.z >= th) ? av.z : 0.0f;
            av.w = (av.w >= th) ? av.w : 0.0f;
            *(v4f*)&Alds[r][c] = av;

            v4f wv = {};
            if (nvalid[i]) {
                wv = __builtin_nontemporal_load((const v4f*)(gW[i] + kt));
                const uint32_t m4 =
                    __builtin_nontemporal_load((const uint32_t*)((const unsigned char*)gM[i] + kt));
                wv.x = (m4 & 0x000000FFu) ? wv.x : 0.0f;
                wv.y = (m4 & 0x0000FF00u) ? wv.y : 0.0f;
                wv.z = (m4 & 0x00FF0000u) ? wv.z : 0.0f;
                wv.w = (m4 & 0xFF000000u) ? wv.w : 0.0f;
            }
            *(v4f*)&Blds[r][c] = wv;

            if (kt + KT < KC) {   // global_prefetch_b8 for next K tile
                __builtin_prefetch(gA[i] + kt + KT, 0, 3);
                __builtin_prefetch(gW[i] + kt + KT, 0, 0);
            }
            (void)arow;
        }
        __syncthreads();

        // ---- 8 k-steps of 4, 8 WMMAs each: D = A(16x4) x B(4x16) + C ----
#pragma unroll
        for (int k4 = 0; k4 < KT; k4 += 4) {
            v2f a[2], b[4];
#pragma unroll
            for (int mi = 0; mi < 2; ++mi)
                a[mi] = *(const v2f*)&Alds[wm * 32 + mi * 16 + l15][k4 + 2 * hi];
#pragma unroll
            for (int ni = 0; ni < 4; ++ni)
                b[ni] = *(const v2f*)&Blds[wn * 64 + ni * 16 + l15][k4 + 2 * hi];
#pragma unroll
            for (int mi = 0; mi < 2; ++mi)
#pragma unroll
                for (int ni = 0; ni < 4; ++ni)
                    acc[mi][ni] = __builtin_amdgcn_wmma_f32_16x16x4_f32(
                        false, a[mi], false, b[ni],
                        (short)0, acc[mi][ni], false, false);
        }
        __syncthreads();
    }

    // ---- write split-K partials (deterministic reduction done separately) ----
    float* pbase = partial + (size_t)ks * ((size_t)B_DIM * NT_PAD);
#pragma unroll
    for (int mi = 0; mi < 2; ++mi) {
#pragma unroll
        for (int ni = 0; ni < 4; ++ni) {
            const int m0 = bm * MT + wm * 32 + mi * 16;
            const int n0 = bn * NTILE + wn * 64 + ni * 16;
#pragma unroll
            for (int j = 0; j < 8; ++j) {
                const int r = m0 + j + 8 * hi;     // C/D layout: vgpr j, lane half
                const int cN = n0 + l15;
                pbase[(size_t)r * NT_PAD + cN] = acc[mi][ni][j];
            }
        }
    }
}

// -------------------------------------------------------------------------
// Kernel 3: fixed-order split-K reduction (deterministic).
// -------------------------------------------------------------------------
__global__ __launch_bounds__(256) void reduce_kernel(
    const float* __restrict__ partial, float* __restrict__ out)
{
    const int idx = blockIdx.x * 256 + threadIdx.x;
    if (idx >= B_DIM * N_CLS) return;
    const int b = idx / N_CLS;
    const int n = idx - b * N_CLS;
    float s = 0.0f;
#pragma unroll
    for (int ksv = 0; ksv < SPLITK; ++ksv)
        s += partial[(size_t)ksv * B_DIM * NT_PAD + (size_t)b * NT_PAD + n];
    out[idx] = s;
}

// -------------------------------------------------------------------------
extern "C" void kernel_launch(void* const* d_in, const int* in_sizes, int n_in,
                              void* d_out, int out_size, void* d_ws, size_t ws_size,
                              hipStream_t stream) {
    (void)in_sizes; (void)n_in; (void)out_size; (void)ws_size;
    const float*         x  = (const float*)d_in[0];
    const float*         wt = (const float*)d_in[1];
    const unsigned char* wm = (const unsigned char*)d_in[2];   // jnp bool = 1 byte
    float* out = (float*)d_out;

    float* thresh  = (float*)d_ws;                              // 256 floats
    float* partial = (float*)((char*)d_ws + 1024);              // 8*256*1024 f32

    topk_thresh_kernel<<<B_DIM, 256, 0, stream>>>(x, thresh);

    dim3 grid(8 /*N tiles*/, 2 /*M tiles*/, SPLITK);
    wmma_gemm_kernel<<<grid, 256, 0, stream>>>(x, wt, wm, thresh, partial);

    const int total = B_DIM * N_CLS;
    reduce_kernel<<<(total + 255) / 256, 256, 0, stream>>>(partial, out);
}